// IngrToInstrAttention_5231270166920
// MI455X (gfx1250) — compile-verified
//
#include <hip/hip_runtime.h>

// Problem dims (fixed by the reference)
#define S_DIM   1024
#define B_DIM   128
#define E_DIM   1024
#define DH_DIM  1024
#define OUT_DIM 1024
#define C_DIM   (E_DIM + DH_DIM)

// Flash-style split of the S dimension
#define NCH     8                    // chunks per batch row
#define SCHUNK  (S_DIM / NCH)        // 128 s-rows per block
#define TS      8                    // s-rows per inner tile

typedef float v2f __attribute__((ext_vector_type(2)));
typedef float v8f __attribute__((ext_vector_type(8)));

// -----------------------------------------------------------------------------
// Wave-level 16x16 f32 tile MAC:  acc += A_tile(16xK) * W_tile(16xK)^T
// ISA fragment layout (32-bit A 16x4 / B 4x16): lane holds the contiguous K-pair
// at [row = lane%16, k + 2*(lane>>4)] -> plain float2 loads for both operands.
// -----------------------------------------------------------------------------
__device__ __forceinline__ v8f wmma_tile_f32(const float* __restrict__ A, int lda,
                                             const float* __restrict__ W, int ldw,
                                             int K, v8f acc, int lr, int half) {
    const float* ap = A + (size_t)lr * lda + 2 * half;
    const float* wp = W + (size_t)lr * ldw + 2 * half;
#pragma unroll 8
    for (int k = 0; k < K; k += 4) {
        v2f a = *(const v2f*)(ap + k);
        v2f b = *(const v2f*)(wp + k);
        acc = __builtin_amdgcn_wmma_f32_16x16x4_f32(false, a, false, b,
                                                    (short)0, acc, false, false);
    }
    return acc;
}

// -----------------------------------------------------------------------------
// K1: dh[b,e] = sum_d hidden[b,d] * Wp[e,d] + bp[e]      (128 x 1024 x 1024)
// One wave per 16x16 output tile; 512 tiles; block = 8 waves.
// -----------------------------------------------------------------------------
__global__ void proj_gemm(const float* __restrict__ hidden,
                          const float* __restrict__ Wp,
                          const float* __restrict__ bp,
                          float* __restrict__ dh) {
    const int wave = threadIdx.x >> 5;
    const int lane = threadIdx.x & 31;
    const int lr   = lane & 15;
    const int half = lane >> 4;
    const int tile = blockIdx.x * (blockDim.x >> 5) + wave;  // 0..511
    const int mtile = (tile & 7) * 16;                        // b
    const int ntile = (tile >> 3) * 16;                       // e

    v8f acc = {};
    acc = wmma_tile_f32(hidden + (size_t)mtile * DH_DIM, DH_DIM,
                        Wp     + (size_t)ntile * DH_DIM, DH_DIM,
                        DH_DIM, acc, lr, half);

    const float bias = bp[ntile + lr];
#pragma unroll
    for (int r = 0; r < 8; ++r)
        dh[(size_t)(mtile + r + 8 * half) * E_DIM + ntile + lr] = acc[r] + bias;
}

// -----------------------------------------------------------------------------
// K2 (fused scores + online softmax + context): single pass over ingr (512 MB).
// Block = (b, s-chunk of 128 rows), 256 threads; thread owns a float4 slice of
// E. Per 8-row tile: coalesced b128 row loads -> partial dots -> LDS/wave
// reduction -> online-softmax rescale + context accumulate from registers.
// Emits per-chunk partials (m, l, ctx[E]) for the combine kernel.
// -----------------------------------------------------------------------------
__global__ void attn_fused_kernel(const float* __restrict__ ingr,
                                  const float* __restrict__ dh,
                                  float* __restrict__ pctx,   // [B][NCH][E]
                                  float* __restrict__ pml) {  // [B][NCH][2]
    __shared__ float part[TS][256];
    __shared__ float sc[TS];

    const int tid  = threadIdx.x;
    const int b    = blockIdx.x >> 3;       // NCH == 8
    const int ch   = blockIdx.x & (NCH - 1);
    const int s0   = ch * SCHUNK;
    const int w    = tid >> 5;
    const int lane = tid & 31;

    // This thread's slice of dh[b,:]
    const float4 q = ((const float4*)(dh + (size_t)b * E_DIM))[tid];

    const float* base = ingr + ((size_t)s0 * B_DIM + b) * E_DIM + 4 * tid;
    const size_t srow = (size_t)B_DIM * E_DIM;

    float4 cacc = make_float4(0.f, 0.f, 0.f, 0.f);
    float  m = -3.0e38f;
    float  l = 0.f;

    for (int t = 0; t < SCHUNK; t += TS) {
        // Load TS rows' float4 slices (stays in registers for the accumulate)
        float4 r[TS];
#pragma unroll
        for (int j = 0; j < TS; ++j)
            r[j] = *(const float4*)(base + (size_t)(t + j) * srow);

        // Partial dot products -> LDS
#pragma unroll
        for (int j = 0; j < TS; ++j)
            part[j][tid] = r[j].x * q.x + r[j].y * q.y + r[j].z * q.z + r[j].w * q.w;
        __syncthreads();

        // Wave w reduces row w (256 partials -> 1 score)
        float p = 0.f;
#pragma unroll
        for (int i = 0; i < 8; ++i)
            p += part[w][lane + 32 * i];
#pragma unroll
        for (int off = 16; off > 0; off >>= 1)
            p += __shfl_xor(p, off, 32);
        if (lane == 0) sc[w] = p;
        __syncthreads();

        // Online softmax update (identical in every thread)
        float tmax = sc[0];
#pragma unroll
        for (int j = 1; j < TS; ++j) tmax = fmaxf(tmax, sc[j]);
        const float nm    = fmaxf(m, tmax);
        const float scale = __expf(m - nm);

        float wgt[TS];
        float lsum = 0.f;
#pragma unroll
        for (int j = 0; j < TS; ++j) {
            wgt[j] = __expf(sc[j] - nm);
            lsum  += wgt[j];
        }
        l = l * scale + lsum;
        cacc.x *= scale; cacc.y *= scale; cacc.z *= scale; cacc.w *= scale;
#pragma unroll
        for (int j = 0; j < TS; ++j) {
            cacc.x += wgt[j] * r[j].x;
            cacc.y += wgt[j] * r[j].y;
            cacc.z += wgt[j] * r[j].z;
            cacc.w += wgt[j] * r[j].w;
        }
        m = nm;
        // No barrier needed here: next tile's part[] writes are fenced by the
        // first __syncthreads, and sc[] reads complete before any thread can
        // pass it.
    }

    ((float4*)(pctx + ((size_t)b * NCH + ch) * E_DIM))[tid] = cacc;
    if (tid == 0) {
        pml[((size_t)b * NCH + ch) * 2 + 0] = m;
        pml[((size_t)b * NCH + ch) * 2 + 1] = l;
    }
}

// -----------------------------------------------------------------------------
// K3: merge the NCH per-chunk partials: ctx[b,e] =
//     sum_c pctx[c][e] * exp(m_c - m*) / sum_c l_c * exp(m_c - m*)
// -----------------------------------------------------------------------------
__global__ void attn_combine_kernel(const float* __restrict__ pctx,
                                    const float* __restrict__ pml,
                                    float* __restrict__ ctx) {
    const int b   = blockIdx.x;
    const int tid = threadIdx.x;

    float mc[NCH], lc[NCH];
    float m = -3.0e38f;
#pragma unroll
    for (int c = 0; c < NCH; ++c) {
        mc[c] = pml[((size_t)b * NCH + c) * 2 + 0];
        lc[c] = pml[((size_t)b * NCH + c) * 2 + 1];
        m = fmaxf(m, mc[c]);
    }
    float a[NCH];
    float L = 0.f;
#pragma unroll
    for (int c = 0; c < NCH; ++c) {
        a[c] = __expf(mc[c] - m);
        L += lc[c] * a[c];
    }
    const float inv = 1.0f / L;

    float4 acc = make_float4(0.f, 0.f, 0.f, 0.f);
#pragma unroll
    for (int c = 0; c < NCH; ++c) {
        const float4 v = ((const float4*)(pctx + ((size_t)b * NCH + c) * E_DIM))[tid];
        acc.x += a[c] * v.x; acc.y += a[c] * v.y;
        acc.z += a[c] * v.z; acc.w += a[c] * v.w;
    }
    acc.x *= inv; acc.y *= inv; acc.z *= inv; acc.w *= inv;
    ((float4*)(ctx + (size_t)b * E_DIM))[tid] = acc;
}

// -----------------------------------------------------------------------------
// K4: out[b,o] = sum_c [ctx|hidden][b,c] * Wout[o,c] + bout[o]
// (128 x 1024 x 2048). K-loop split at the concat boundary; 'concat' never
// materialized. One wave per 16x16 tile; 512 tiles.
// -----------------------------------------------------------------------------
__global__ void out_gemm(const float* __restrict__ ctx,
                         const float* __restrict__ hidden,
                         const float* __restrict__ Wout,
                         const float* __restrict__ bout,
                         float* __restrict__ out) {
    const int wave = threadIdx.x >> 5;
    const int lane = threadIdx.x & 31;
    const int lr   = lane & 15;
    const int half = lane >> 4;
    const int tile = blockIdx.x * (blockDim.x >> 5) + wave;  // 0..511
    const int mtile = (tile & 7) * 16;                        // b
    const int ntile = (tile >> 3) * 16;                       // o

    v8f acc = {};
    acc = wmma_tile_f32(ctx  + (size_t)mtile * E_DIM, E_DIM,
                        Wout + (size_t)ntile * C_DIM, C_DIM,
                        E_DIM, acc, lr, half);
    acc = wmma_tile_f32(hidden + (size_t)mtile * DH_DIM, DH_DIM,
                        Wout   + (size_t)ntile * C_DIM + E_DIM, C_DIM,
                        DH_DIM, acc, lr, half);

    const float bias = bout[ntile + lr];
#pragma unroll
    for (int r = 0; r < 8; ++r)
        out[(size_t)(mtile + r + 8 * half) * OUT_DIM + ntile + lr] = acc[r] + bias;
}

// -----------------------------------------------------------------------------
extern "C" void kernel_launch(void* const* d_in, const int* in_sizes, int n_in,
                              void* d_out, int out_size, void* d_ws, size_t ws_size,
                              hipStream_t stream) {
    const float* ingr   = (const float*)d_in[0];  // [S,B,E]
    const float* hidden = (const float*)d_in[1];  // [1,B,DH]
    const float* Wp     = (const float*)d_in[2];  // [E,DH]
    const float* bp     = (const float*)d_in[3];  // [E]
    const float* Wout   = (const float*)d_in[4];  // [OUT,E+DH]
    const float* bout   = (const float*)d_in[5];  // [OUT]
    float* out = (float*)d_out;                   // [1,B,OUT]

    float* dh   = (float*)d_ws;                          // [B,E]        512 KB
    float* pctx = dh + (size_t)B_DIM * E_DIM;            // [B,NCH,E]    4 MB
    float* pml  = pctx + (size_t)B_DIM * NCH * E_DIM;    // [B,NCH,2]    8 KB
    float* ctx  = pml + (size_t)B_DIM * NCH * 2;         // [B,E]        512 KB

    proj_gemm<<<(B_DIM / 16) * (E_DIM / 16) / 8, 256, 0, stream>>>(hidden, Wp, bp, dh);
    attn_fused_kernel<<<B_DIM * NCH, 256, 0, stream>>>(ingr, dh, pctx, pml);
    attn_combine_kernel<<<B_DIM, 256, 0, stream>>>(pctx, pml, ctx);
    out_gemm<<<(B_DIM / 16) * (OUT_DIM / 16) / 8, 256, 0, stream>>>(ctx, hidden, Wout, bout, out);
}